// Pooling_46634754900485
// MI455X (gfx1250) — compile-verified
//
#include <hip/hip_runtime.h>

typedef __attribute__((ext_vector_type(2))) float v2f;
typedef __attribute__((ext_vector_type(4))) float fvec4;
typedef __attribute__((ext_vector_type(8))) float v8f;
typedef int v4i __attribute__((vector_size(16)));

#define T_ 1024
#define D_ 1024
#define B_ 4
#define TT 64              // t-tile per workgroup
#define DT 128             // d-tile per workgroup (8 waves x 16)
#define DTP 144            // LDS row pitch (floats): 2-row shift -> 32-bank shift, conflict-free
#define ROWS (TT + 16)     // halo rows t0-7 .. t0+TT+8
#define TB4D ((size_t)T_ * B_ * 4 * D_)   // per-scale-group output size (floats)
#define ROWSTRIDE ((size_t)(B_ * 4 * D_)) // output stride per t (floats)

#if __has_builtin(__builtin_amdgcn_global_load_async_to_lds_b128) && \
    __has_builtin(__builtin_amdgcn_s_wait_asynccnt)
#define USE_ASYNC_LDS 1
#endif

// Combined Toeplitz weight: Wc[m][k], m=0..15 (12 used), k=0..15 taps of x[t-7+k].
__device__ __forceinline__ float wc_val(int m, int k,
                                        const float* W4, const float* W8, const float* W16) {
  if (m < 4)  return (k >= 6 && k < 10) ? W4[m * 4 + (k - 6)]       : 0.0f;
  if (m < 8)  return (k >= 4 && k < 12) ? W8[(m - 4) * 8 + (k - 4)] : 0.0f;
  if (m < 12) return W16[(m - 8) * 16 + k];
  return 0.0f;
}

__global__ __launch_bounds__(256)
void msconv_wmma_main(const float* __restrict__ x,
                      const float* __restrict__ W4, const float* __restrict__ W8,
                      const float* __restrict__ W16, float* __restrict__ out) {
  __shared__ float lds[ROWS * DTP];
  const int tid = threadIdx.x;
  const int d0  = blockIdx.x * DT;
  const int t0  = blockIdx.y * TT;
  const int b   = blockIdx.z;
  const float* xb = x + (size_t)b * T_ * D_;

  // ---- Stage x tile: rows [t0-7, t0+TT+8] (edge-clamped), cols [d0, d0+DT) ----
  for (int li = tid; li < ROWS * (DT / 4); li += 256) {
    int row = li >> 5;            // DT/4 = 32 float4 per row
    int c4  = li & 31;
    int gr  = t0 - 7 + row;
    gr = gr < 0 ? 0 : (gr > T_ - 1 ? T_ - 1 : gr);
    const float* gp = xb + (size_t)gr * D_ + d0 + (c4 << 2);
    float* lp = &lds[row * DTP + (c4 << 2)];
#ifdef USE_ASYNC_LDS
    __builtin_amdgcn_global_load_async_to_lds_b128(
        (__attribute__((address_space(1))) v4i*)gp,
        (__attribute__((address_space(3))) v4i*)lp, 0, 0);
#else
    *(fvec4*)lp = *(const fvec4*)gp;
#endif
  }
#ifdef USE_ASYNC_LDS
  __builtin_amdgcn_s_wait_asynccnt(0);
#endif
  __syncthreads();

  const int lane = tid & 31;
  const int wv   = tid >> 5;     // wave id -> 16-d sub-column
  const int h    = lane >> 4;    // lane half
  const int n    = lane & 15;

  // ---- A fragments: Wc in 16x4 f32 A layout, (VGPR j, lane L) = A[M=L%16][K=2h+j] ----
  v2f a[4];
#pragma unroll
  for (int c = 0; c < 4; ++c) {
    int k0 = 4 * c + 2 * h;
    a[c][0] = wc_val(n, k0,     W4, W8, W16);
    a[c][1] = wc_val(n, k0 + 1, W4, W8, W16);
  }

  // ---- Per-lane output base pointers (m = r + 8h -> scale group g, channel o) ----
  const int dcol = d0 + wv * 16 + n;
  float* pr[8];
#pragma unroll
  for (int r = 0; r < 8; ++r) {
    int mm = r + 8 * h;
    int g = mm >> 2, o = mm & 3;
    if (mm >= 12) { g = 0; o = 0; }   // dead lanes; store is masked below
    pr[r] = out + (size_t)g * TB4D + (size_t)t0 * ROWSTRIDE +
            (size_t)(b * 4 + o) * D_ + dcol;
  }

  // ---- Sweep t: B[k][n] = x[t-7+k, dcol]; chunk c rows (tl + 4c + 2h + j) ----
  const int lbase = wv * 16 + n;
  for (int tl = 0; tl < TT; ++tl) {
    const int rb = (tl + 2 * h) * DTP + lbase;
    v2f bb[4];
#pragma unroll
    for (int c = 0; c < 4; ++c) {
      bb[c][0] = lds[rb + (4 * c)     * DTP];
      bb[c][1] = lds[rb + (4 * c + 1) * DTP];
    }
    v8f acc = {0.f, 0.f, 0.f, 0.f, 0.f, 0.f, 0.f, 0.f};
#pragma unroll
    for (int c = 0; c < 4; ++c)
      acc = __builtin_amdgcn_wmma_f32_16x16x4_f32(false, a[c], false, bb[c],
                                                  (short)0, acc, false, false);
    const size_t toff = (size_t)tl * ROWSTRIDE;
#pragma unroll
    for (int r = 0; r < 8; ++r) {
      if (r < 4 || h == 0) pr[r][toff] = acc[r];   // m = r+8h valid iff < 12
    }
  }
}

// Boundary t's (0..6 and 1016..1023): per-scale clipped/shifted windows; overwrites main.
__global__ void msconv_boundary(const float* __restrict__ x,
                                const float* __restrict__ W4, const float* __restrict__ W8,
                                const float* __restrict__ W16, float* __restrict__ out) {
  int gid = blockIdx.x * blockDim.x + threadIdx.x;
  if (gid >= 15 * B_ * D_) return;
  int d  = gid % D_;
  int b  = (gid / D_) % B_;
  int ti = gid / (D_ * B_);
  int t  = (ti < 7) ? ti : (T_ - 8 + (ti - 7));
  const float* xb = x + (size_t)b * T_ * D_;
  const float* Ws[3] = {W4, W8, W16};
  const int ss[3] = {4, 8, 16};
#pragma unroll
  for (int g = 0; g < 3; ++g) {
    const int s = ss[g];
    const float* W = Ws[g];
    int st = t - s / 2 + 1; if (st < 0) st = 0;
    int la = t + s / 2;     if (la > T_ - 1) la = T_ - 1;
    float a0 = 0.f, a1 = 0.f, a2 = 0.f, a3 = 0.f;
    for (int k = 0; k < s; ++k) {
      int u = st + k;
      if (u > la) break;
      float xv = xb[(size_t)u * D_ + d];
      a0 += W[0 * s + k] * xv; a1 += W[1 * s + k] * xv;
      a2 += W[2 * s + k] * xv; a3 += W[3 * s + k] * xv;
    }
    size_t base = (size_t)g * TB4D + ((size_t)t * B_ + b) * 4 * D_ + d;
    out[base] = a0; out[base + D_] = a1; out[base + 2 * D_] = a2; out[base + 3 * D_] = a3;
  }
}

extern "C" void kernel_launch(void* const* d_in, const int* in_sizes, int n_in,
                              void* d_out, int out_size, void* d_ws, size_t ws_size,
                              hipStream_t stream) {
  const float* x   = (const float*)d_in[0];
  const float* W4  = (const float*)d_in[1];
  const float* W8  = (const float*)d_in[2];
  const float* W16 = (const float*)d_in[3];
  float* out = (float*)d_out;

  dim3 grid(D_ / DT, T_ / TT, B_);           // 8 x 16 x 4 = 512 workgroups
  msconv_wmma_main<<<grid, 256, 0, stream>>>(x, W4, W8, W16, out);

  int nthreads = 15 * B_ * D_;
  msconv_boundary<<<(nthreads + 255) / 256, 256, 0, stream>>>(x, W4, W8, W16, out);
}